// InvariancePropagationLoss_1752346657500
// MI455X (gfx1250) — compile-verified
//
#include <hip/hip_runtime.h>
#include <hip/hip_bf16.h>
#include <math.h>

// ---------------- problem constants (match reference) ----------------
#define BB      128          // batch
#define DD      128          // feature dim
#define NN      1000000      // bank size
#define KNEI    4
#define NNEI    84           // propagate() output width: 4 + 20*4
#define NPOS    50
#define NBG     4096
#define T_INV   (1.0f / 0.07f)

// ---------------- selection constants ----------------
#define NBINS   2048
#define CAP     6144         // per-row candidate cap (expected ~4.3K)
#define SORTN   8192         // bitonic size (pow2 >= CAP)

// ---------------- GEMM tiling ----------------
#define TILE_N  64
#define LDSB_STRIDE 132      // pad: 132 % 64 = 4 -> conflict-free column reads
                             // and rows stay 16B-aligned for async b128 LDS writes

typedef __attribute__((ext_vector_type(2))) float v2f;
typedef __attribute__((ext_vector_type(8))) float v8f;

// LDS byte offset of a pointer into a __shared__ array (generic -> as(3) -> u32)
typedef __attribute__((address_space(3))) void lds_void;
__device__ __forceinline__ unsigned lds_off(const void* p) {
  return (unsigned)(unsigned long long)(lds_void*)p;
}

// =====================================================================
// 0) init: zero histogram + candidate counters
// =====================================================================
__global__ void k_init(unsigned* __restrict__ hist, unsigned* __restrict__ ccount) {
  int i = blockIdx.x * blockDim.x + threadIdx.x;
  if (i < BB * NBINS) hist[i] = 0u;
  if (i < BB) ccount[i] = 0u;
}

// =====================================================================
// 1) normalize points (one block per row)
// =====================================================================
__global__ __launch_bounds__(DD) void k_normalize(const float* __restrict__ pts,
                                                  float* __restrict__ normP) {
  __shared__ float red[DD];
  const int b = blockIdx.x, t = threadIdx.x;
  const float x = pts[b * DD + t];
  red[t] = x * x;
  __syncthreads();
  for (int s = DD / 2; s > 0; s >>= 1) {
    if (t < s) red[t] += red[t + s];
    __syncthreads();
  }
  normP[b * DD + t] = x * (1.0f / sqrtf(red[0]));
}

// =====================================================================
// 2) neighbor propagation (reference: 4 seed cols + 20 expansion steps)
// =====================================================================
__global__ void k_propagate(const int* __restrict__ pidx, const int* __restrict__ neigh,
                            int* __restrict__ neighs_ws, int* __restrict__ out_neighs) {
  const int b = threadIdx.x;
  if (b >= BB) return;
  int cols[NNEI];
  const int p = pidx[b];
  for (int j = 0; j < KNEI; ++j) cols[j] = neigh[(size_t)p * KNEI + j];
  for (int s = 0; s < 20; ++s) {
    const int cur = cols[s];
    for (int j = 0; j < KNEI; ++j) cols[4 + 4 * s + j] = neigh[(size_t)cur * KNEI + j];
  }
  for (int j = 0; j < NNEI; ++j) {
    neighs_ws[b * NNEI + j] = cols[j];
    out_neighs[b * NNEI + j] = cols[j];
  }
}

// =====================================================================
// async stage: 64 bank rows (32 KB) -> LDS buffer, 8 x b128 per thread,
// direct global->LDS via GLOBAL_LOAD_ASYNC_TO_LDS_B128 (ASYNCcnt)
// =====================================================================
__device__ __forceinline__ void stage_async(const float* __restrict__ bank,
                                            int ncolbase, float* dstbuf, int tid) {
#pragma unroll
  for (int c = 0; c < 8; ++c) {
    const int idx = tid + c * 256;          // 0..2047
    const int n  = idx >> 5;                // bank row within tile: 0..63
    const int f4 = idx & 31;                // 16B chunk within row: 0..31
    const unsigned goff = (unsigned)(((size_t)(ncolbase + n) * DD + (size_t)f4 * 4)
                                     * sizeof(float));
    const unsigned la = lds_off(&dstbuf[n * LDSB_STRIDE + f4 * 4]);
    asm volatile("global_load_async_to_lds_b128 %0, %1, %2"
                 :: "v"(la), "v"(goff), "s"(bank) : "memory");
  }
}

// =====================================================================
// 3+5) fused WMMA GEMM: sim = normP @ bank^T  (fp32, 16x16x4 WMMA)
//   double-buffered LDS bank tiles, async global->LDS staging
//   MODE 0: per-row value histogram (threshold finding)
//   MODE 1: collect candidates >= per-row threshold bin
// =====================================================================
template <int MODE>
__global__ __launch_bounds__(256) void k_gemm(const float* __restrict__ normP,
                                              const float* __restrict__ bank,
                                              unsigned* __restrict__ hist,
                                              const int* __restrict__ tbin,
                                              unsigned* __restrict__ ccount,
                                              float* __restrict__ cval,
                                              int* __restrict__ cidx,
                                              int ntiles) {
  __shared__ float ldsB[2][TILE_N * LDSB_STRIDE];
  const int tid  = threadIdx.x;
  const int wave = tid >> 5;
  const int lane = tid & 31;
  const int lo   = lane & 15;       // M (for A/C) or N (for B/C column)
  const int hi   = lane >> 4;       // half-select per ISA f32 layouts
  const int rowbase = wave * 16;    // each wave owns 16 of the 128 point rows

  // ---- A fragments (this wave's 16 rows, all K=128), loaded once ----
  // f32 A 16x4 layout: lanes 0-15 hold K={k,k+1}, lanes 16-31 hold K={k+2,k+3}
  v2f areg[32];
#pragma unroll
  for (int kk = 0; kk < 32; ++kk) {
    const int kb = kk * 4 + 2 * hi;
    const float2 a2 = *(const float2*)(normP + (size_t)(rowbase + lo) * DD + kb);
    v2f av = {a2.x, a2.y};
    areg[kk] = av;
  }

  int tile = blockIdx.x;
  if (tile < ntiles) stage_async(bank, tile * TILE_N, ldsB[0], tid);
  int buf = 0;

  for (; tile < ntiles; tile += gridDim.x) {
    const int nxt = tile + (int)gridDim.x;
    const bool have_nxt = nxt < ntiles;
    // prefetch-issue the following tile into the other buffer (its readers
    // finished at the previous iteration's trailing barrier)
    if (have_nxt) {
      stage_async(bank, nxt * TILE_N, ldsB[buf ^ 1], tid);
      // deep prefetch two tiles ahead into L2
      if (nxt + (int)gridDim.x < ntiles) {
        const float* far = bank + (size_t)(nxt + gridDim.x) * TILE_N * DD;
        __builtin_prefetch(far + (size_t)tid * 32, 0, 1);
      }
      // async loads complete in order: first 8 (current tile) done at cnt<=8
      asm volatile("s_wait_asynccnt 0x8" ::: "memory");
    } else {
      asm volatile("s_wait_asynccnt 0x0" ::: "memory");
    }
    __syncthreads();

    const float* curB = ldsB[buf];
    const int ncolbase = tile * TILE_N;

    // ---- 4 subtiles of 16 bank columns each ----
    for (int t = 0; t < 4; ++t) {
      v8f acc = {};
      const float* bcol = &curB[(t * 16 + lo) * LDSB_STRIDE + 2 * hi];
#pragma unroll
      for (int kk = 0; kk < 32; ++kk) {
        const float2 b2 = *(const float2*)(bcol + kk * 4);
        v2f bfrag = {b2.x, b2.y};
        acc = __builtin_amdgcn_wmma_f32_16x16x4_f32(
            /*neg_a=*/false, areg[kk], /*neg_b=*/false, bfrag,
            /*c_mod=*/(short)0, acc, /*reuse_a=*/false, /*reuse_b=*/false);
      }
      const int ncol = ncolbase + t * 16 + lo;   // this lane's bank index
#pragma unroll
      for (int j = 0; j < 8; ++j) {
        const float v = acc[j];
        const int row = rowbase + j + 8 * hi;    // C layout: M = j + 8*hi
        int bin = (int)((v + 1.0f) * (NBINS * 0.5f));
        bin = bin < 0 ? 0 : (bin > NBINS - 1 ? NBINS - 1 : bin);
        if constexpr (MODE == 0) {
          atomicAdd(&hist[row * NBINS + bin], 1u);
        } else {
          if (bin >= tbin[row]) {
            const unsigned p = atomicAdd(&ccount[row], 1u);
            if (p < CAP) {
              cval[(size_t)row * CAP + p] = v;
              cidx[(size_t)row * CAP + p] = ncol;
            }
          }
        }
      }
    }
    __syncthreads();   // all waves done reading ldsB[buf] before it is re-staged
    buf ^= 1;
  }
}

// =====================================================================
// 4) per-row threshold bin: largest bin s.t. count(>=bin) >= NBG
// =====================================================================
__global__ void k_threshold(const unsigned* __restrict__ hist, int* __restrict__ tbin) {
  const int b = threadIdx.x;
  if (b >= BB) return;
  unsigned cum = 0;
  int t = 0;
  for (int bin = NBINS - 1; bin >= 0; --bin) {
    cum += hist[b * NBINS + bin];
    if (cum >= NBG) { t = bin; break; }
  }
  tbin[b] = t;
}

// =====================================================================
// 6) per-row bitonic sort (descending) of candidates; emit top-4096
//    indices + bg_sum = sum exp(v/T)
// =====================================================================
__global__ __launch_bounds__(1024) void k_sort_bg(const float* __restrict__ cval,
                                                  const int* __restrict__ cidx,
                                                  const unsigned* __restrict__ ccount,
                                                  int* __restrict__ out_idx,
                                                  float* __restrict__ bgsum) {
  extern __shared__ char smem[];
  float* sv = (float*)smem;
  int*   si = (int*)(smem + SORTN * sizeof(float));
  const int b = blockIdx.x, t = threadIdx.x;
  int cnt = (int)ccount[b];
  if (cnt > CAP) cnt = CAP;
  for (int i = t; i < SORTN; i += 1024) {
    if (i < cnt) { sv[i] = cval[(size_t)b * CAP + i]; si[i] = cidx[(size_t)b * CAP + i]; }
    else         { sv[i] = -3.0f;                     si[i] = -1; }
  }
  __syncthreads();
  for (int k = 2; k <= SORTN; k <<= 1) {
    for (int j = k >> 1; j > 0; j >>= 1) {
      for (int i = t; i < SORTN; i += 1024) {
        const int l = i ^ j;
        if (l > i) {
          const bool up = ((i & k) == 0);
          const float a = sv[i], c = sv[l];
          if ((a < c) == up) {            // descending sort
            sv[i] = c; sv[l] = a;
            const int ti = si[i]; si[i] = si[l]; si[l] = ti;
          }
        }
      }
      __syncthreads();
    }
  }
  float partial = 0.f;
  for (int i = t; i < NBG; i += 1024) {
    out_idx[(size_t)b * NBG + i] = si[i];
    partial += __expf(sv[i] * T_INV);
  }
  __syncthreads();
  sv[t] = partial;
  __syncthreads();
  for (int s = 512; s > 0; s >>= 1) {
    if (t < s) sv[t] += sv[t + s];
    __syncthreads();
  }
  if (t == 0) bgsum[b] = sv[0];
}

// =====================================================================
// 7) self similarity + 84 positive similarities (exp), recomputed dots
// =====================================================================
__global__ __launch_bounds__(256) void k_selfpos(const float* __restrict__ normP,
                                                 const float* __restrict__ bank,
                                                 const int* __restrict__ pidx,
                                                 const int* __restrict__ neighs,
                                                 float* __restrict__ posext,
                                                 float* __restrict__ selfv) {
  const int b = blockIdx.x;
  const int wave = threadIdx.x >> 5, lane = threadIdx.x & 31;
  for (int j = wave; j < NNEI + 1; j += 8) {
    const int idx = (j < NNEI) ? neighs[b * NNEI + j] : pidx[b];
    const float4 pv = ((const float4*)(normP + (size_t)b * DD))[lane];
    const float4 bv = ((const float4*)(bank + (size_t)idx * DD))[lane];
    float s = pv.x * bv.x + pv.y * bv.y + pv.z * bv.z + pv.w * bv.w;
    for (int off = 16; off > 0; off >>= 1) s += __shfl_xor(s, off);
    if (lane == 0) {
      const float e = __expf(s * T_INV);
      if (j < NNEI) posext[b * NNEI + j] = e;
      else          selfv[b] = e;
    }
  }
}

// =====================================================================
// 8) hard positives: sum of the 50 smallest of 84 positive sims
// =====================================================================
__global__ __launch_bounds__(128) void k_hardpos(const float* __restrict__ posext,
                                                 float* __restrict__ hard) {
  __shared__ float pv[128];
  __shared__ float red[128];
  const int b = blockIdx.x, t = threadIdx.x;
  pv[t] = (t < NNEI) ? posext[b * NNEI + t] : 3.4e38f;
  __syncthreads();
  for (int k = 2; k <= 128; k <<= 1) {
    for (int j = k >> 1; j > 0; j >>= 1) {
      const int l = t ^ j;
      if (l > t) {
        const bool up = ((t & k) == 0);
        const float a = pv[t], c = pv[l];
        if ((a > c) == up) { pv[t] = c; pv[l] = a; }   // ascending sort
      }
      __syncthreads();
    }
  }
  red[t] = (t < NPOS) ? pv[t] : 0.f;
  __syncthreads();
  for (int s = 64; s > 0; s >>= 1) {
    if (t < s) red[t] += red[t + s];
    __syncthreads();
  }
  if (t == 0) hard[b] = red[0];
}

// =====================================================================
// 9) final losses
// =====================================================================
__global__ __launch_bounds__(128) void k_loss(const float* __restrict__ selfv,
                                              const float* __restrict__ bgsum,
                                              const float* __restrict__ hard,
                                              float* __restrict__ out) {
  __shared__ float ra[128], rb[128];
  const int t = threadIdx.x;
  const float sv = selfv[t], bg = bgsum[t], hd = hard[t];
  ra[t] = logf(sv / bg + 1e-7f);
  rb[t] = logf(hd / (bg - sv) + 1e-7f);
  __syncthreads();
  for (int s = 64; s > 0; s >>= 1) {
    if (t < s) { ra[t] += ra[t + s]; rb[t] += rb[t + s]; }
    __syncthreads();
  }
  if (t == 0) { out[0] = -ra[0] / (float)BB; out[1] = -rb[0] / (float)BB; }
}

// =====================================================================
// launcher
// =====================================================================
extern "C" void kernel_launch(void* const* d_in, const int* in_sizes, int n_in,
                              void* d_out, int out_size, void* d_ws, size_t ws_size,
                              hipStream_t stream) {
  const float* points = (const float*)d_in[0];
  const int*   pidx   = (const int*)  d_in[1];
  const float* bank   = (const float*)d_in[2];
  const int*   neigh  = (const int*)  d_in[3];
  (void)in_sizes; (void)n_in; (void)out_size; (void)ws_size;

  // ---- workspace carve-up (all offsets 256B aligned) ----
  char* ws = (char*)d_ws;
  size_t off = 0;
  auto alloc = [&](size_t bytes) { char* p = ws + off; off = (off + bytes + 255) & ~(size_t)255; return p; };
  float*    normP   = (float*)   alloc((size_t)BB * DD * 4);
  unsigned* hist    = (unsigned*)alloc((size_t)BB * NBINS * 4);
  int*      tbin    = (int*)     alloc((size_t)BB * 4);
  unsigned* ccount  = (unsigned*)alloc((size_t)BB * 4);
  float*    cval    = (float*)   alloc((size_t)BB * CAP * 4);
  int*      cidx    = (int*)     alloc((size_t)BB * CAP * 4);
  int*      neighsW = (int*)     alloc((size_t)BB * NNEI * 4);
  float*    posext  = (float*)   alloc((size_t)BB * NNEI * 4);
  float*    selfv   = (float*)   alloc((size_t)BB * 4);
  float*    bgsum   = (float*)   alloc((size_t)BB * 4);
  float*    hard    = (float*)   alloc((size_t)BB * 4);

  // ---- output layout: [lossA, lossB, bg_indices(128*4096), neighs(128*84)] ----
  float* outf      = (float*)d_out;
  int*   out_bgidx = ((int*)d_out) + 2;
  int*   out_neigh = ((int*)d_out) + 2 + (size_t)BB * NBG;

  const int ntiles = NN / TILE_N;   // 15625

  k_init      <<<(BB * NBINS + 255) / 256, 256, 0, stream>>>(hist, ccount);
  k_normalize <<<BB, DD, 0, stream>>>(points, normP);
  k_propagate <<<1, BB, 0, stream>>>(pidx, neigh, neighsW, out_neigh);
  k_gemm<0>   <<<2048, 256, 0, stream>>>(normP, bank, hist, tbin, ccount, cval, cidx, ntiles);
  k_threshold <<<1, BB, 0, stream>>>(hist, tbin);
  k_gemm<1>   <<<2048, 256, 0, stream>>>(normP, bank, hist, tbin, ccount, cval, cidx, ntiles);
  k_sort_bg   <<<BB, 1024, SORTN * 8, stream>>>(cval, cidx, ccount, out_bgidx, bgsum);
  k_selfpos   <<<BB, 256, 0, stream>>>(normP, bank, pidx, neighsW, posext, selfv);
  k_hardpos   <<<BB, 128, 0, stream>>>(posext, hard);
  k_loss      <<<1, BB, 0, stream>>>(selfv, bgsum, hard, outf);
}